// GNNEncoder_37237366456878
// MI455X (gfx1250) — compile-verified
//
#include <hip/hip_runtime.h>

#define N_NODES  50000
#define N_EDGES  600000
#define DIM      128
#define N_GRAPHS 512
#define PAD      132   // padded LDS row stride (floats) -> bank-conflict free

typedef __attribute__((ext_vector_type(2))) float v2f;
typedef __attribute__((ext_vector_type(8))) float v8f;

// ---------------------------------------------------------------- zero
__global__ void zero_f32(float* __restrict__ p, long n) {
  long i = (long)blockIdx.x * blockDim.x + threadIdx.x;
  if (i < n) p[i] = 0.0f;
}

// ---------------------------------------------------------------- edge scatter: agg[dst] += x[src]
__global__ void gin_scatter(const float* __restrict__ xin,
                            const int* __restrict__ src,
                            const int* __restrict__ dst,
                            float* __restrict__ agg) {
  long t = (long)blockIdx.x * blockDim.x + threadIdx.x;
  long e = t >> 5;            // edge id
  int  g = (int)(t & 31);     // 4-channel group 0..31
  if (e >= N_EDGES) return;
  int s = src[e], d = dst[e];
  const float4 v = ((const float4*)(xin + (long)s * DIM))[g];
  float* base = agg + (long)d * DIM + g * 4;
  atomicAdd(base + 0, v.x);
  atomicAdd(base + 1, v.y);
  atomicAdd(base + 2, v.z);
  atomicAdd(base + 3, v.w);
}

// ---------------------------------------------------------------- fused GIN MLP via fp32 WMMA
// block = 256 threads = 8 waves; 16 nodes per block; wave w owns cols [16w,16w+16)
__global__ __launch_bounds__(256) void gin_mlp_wmma(
    const float* __restrict__ xin, const float* __restrict__ agg,
    const float* __restrict__ W1, const float* __restrict__ b1,
    const float* __restrict__ W2, const float* __restrict__ b2,
    float* __restrict__ xout) {
  __shared__ float sW[DIM * PAD];   // weight matrix (reused for W1 then W2)
  __shared__ float sH[16 * PAD];    // input tile  h = x + agg   [16 x 128]
  __shared__ float sT[16 * PAD];    // intermediate relu(h@W1+b1) [16 x 128]

  const int t    = threadIdx.x;
  const int lane = t & 31;
  const int wave = t >> 5;      // 0..7
  const int hi   = lane >> 4;   // half-wave select
  const int lm   = lane & 15;
  const int n0   = wave * 16;
  const long mbase = (long)blockIdx.x * 16;

  // load H tile: h = x + agg  (16 rows x 32 float4)
  for (int i = t; i < 16 * 32; i += 256) {
    int r = i >> 5, c4 = i & 31;
    float4 a = ((const float4*)(xin + (mbase + r) * DIM))[c4];
    float4 b = ((const float4*)(agg + (mbase + r) * DIM))[c4];
    float4 s; s.x = a.x + b.x; s.y = a.y + b.y; s.z = a.z + b.z; s.w = a.w + b.w;
    *(float4*)(&sH[r * PAD + c4 * 4]) = s;
  }
  // load W1 (128 x 32 float4)
  for (int i = t; i < DIM * 32; i += 256) {
    int r = i >> 5, c4 = i & 31;
    *(float4*)(&sW[r * PAD + c4 * 4]) = ((const float4*)(W1 + (long)r * DIM))[c4];
  }
  __syncthreads();

  // ---- GEMM1: T = relu(H @ W1 + b1), wave tile [16 x 16] at cols n0
  v8f acc = {};
#pragma unroll
  for (int k0 = 0; k0 < DIM; k0 += 4) {
    v2f a, b;
    // A frag: lane L, vgpr v -> A[L&15][k0 + v + 2*(L>>4)]
    a.x = sH[lm * PAD + k0 + 2 * hi];
    a.y = sH[lm * PAD + k0 + 2 * hi + 1];
    // B frag: lane L, vgpr v -> B[k0 + v + 2*(L>>4)][n0 + (L&15)]
    b.x = sW[(k0 + 2 * hi) * PAD + n0 + lm];
    b.y = sW[(k0 + 2 * hi + 1) * PAD + n0 + lm];
    acc = __builtin_amdgcn_wmma_f32_16x16x4_f32(false, a, false, b,
                                                (short)0, acc, false, false);
  }
  {
    float bias = b1[n0 + lm];
#pragma unroll
    for (int r = 0; r < 8; ++r) {
      int m = r + 8 * hi;   // C/D layout: vgpr r -> row r (+8 for upper half-wave)
      sT[m * PAD + n0 + lm] = fmaxf(acc[r] + bias, 0.0f);
    }
  }
  __syncthreads();

  // overwrite sW with W2
  for (int i = t; i < DIM * 32; i += 256) {
    int r = i >> 5, c4 = i & 31;
    *(float4*)(&sW[r * PAD + c4 * 4]) = ((const float4*)(W2 + (long)r * DIM))[c4];
  }
  __syncthreads();

  // ---- GEMM2: out = relu(T @ W2 + b2)
  acc = (v8f){};
#pragma unroll
  for (int k0 = 0; k0 < DIM; k0 += 4) {
    v2f a, b;
    a.x = sT[lm * PAD + k0 + 2 * hi];
    a.y = sT[lm * PAD + k0 + 2 * hi + 1];
    b.x = sW[(k0 + 2 * hi) * PAD + n0 + lm];
    b.y = sW[(k0 + 2 * hi + 1) * PAD + n0 + lm];
    acc = __builtin_amdgcn_wmma_f32_16x16x4_f32(false, a, false, b,
                                                (short)0, acc, false, false);
  }
  {
    float bias = b2[n0 + lm];
#pragma unroll
    for (int r = 0; r < 8; ++r) {
      int m = r + 8 * hi;
      xout[(mbase + m) * DIM + n0 + lm] = fmaxf(acc[r] + bias, 0.0f);
    }
  }
}

// ---------------------------------------------------------------- mean pool (scatter)
__global__ void pool_scatter(const float* __restrict__ x,
                             const int* __restrict__ batch,
                             float* __restrict__ sums,
                             float* __restrict__ cnts) {
  long t = (long)blockIdx.x * blockDim.x + threadIdx.x;
  long node = t >> 5;
  int  g = (int)(t & 31);
  if (node >= N_NODES) return;
  int b = batch[node];
  const float4 v = ((const float4*)(x + node * DIM))[g];
  float* base = sums + (long)b * DIM + g * 4;
  atomicAdd(base + 0, v.x);
  atomicAdd(base + 1, v.y);
  atomicAdd(base + 2, v.z);
  atomicAdd(base + 3, v.w);
  if (g == 0) atomicAdd(&cnts[b], 1.0f);
}

__global__ void pool_div(const float* __restrict__ sums,
                         const float* __restrict__ cnts,
                         float* __restrict__ out) {
  int i = blockIdx.x * blockDim.x + threadIdx.x;
  if (i < N_GRAPHS * DIM) out[i] = sums[i] / fmaxf(cnts[i >> 7], 1.0f);
}

// ---------------------------------------------------------------- driver
extern "C" void kernel_launch(void* const* d_in, const int* in_sizes, int n_in,
                              void* d_out, int out_size, void* d_ws, size_t ws_size,
                              hipStream_t stream) {
  // input order: x, (W1,b1,W2,b2) x3, edge_index, batch
  const float* x    = (const float*)d_in[0];
  const int* edge   = (const int*)d_in[13];
  const int* batch  = (const int*)d_in[14];
  const int* src    = edge;
  const int* dst    = edge + N_EDGES;

  float* ws   = (float*)d_ws;
  const long nf = (long)N_NODES * DIM;
  float* agg  = ws;
  float* bufA = ws + nf;
  float* bufB = bufA + nf;
  float* sums = bufB + nf;
  float* cnts = sums + (long)N_GRAPHS * DIM;

  const float* xin = x;
  float* bufs[2] = {bufA, bufB};
  for (int l = 0; l < 3; ++l) {
    const float* W1 = (const float*)d_in[1 + 4 * l + 0];
    const float* b1 = (const float*)d_in[1 + 4 * l + 1];
    const float* W2 = (const float*)d_in[1 + 4 * l + 2];
    const float* b2 = (const float*)d_in[1 + 4 * l + 3];
    float* xout = bufs[l & 1];

    zero_f32<<<(int)((nf + 255) / 256), 256, 0, stream>>>(agg, nf);
    gin_scatter<<<(N_EDGES * 32 + 255) / 256, 256, 0, stream>>>(xin, src, dst, agg);
    gin_mlp_wmma<<<N_NODES / 16, 256, 0, stream>>>(xin, agg, W1, b1, W2, b2, xout);
    xin = xout;
  }

  long npool = (long)N_GRAPHS * DIM + N_GRAPHS;  // sums + cnts contiguous
  zero_f32<<<(int)((npool + 255) / 256), 256, 0, stream>>>(sums, npool);
  pool_scatter<<<(N_NODES * 32 + 255) / 256, 256, 0, stream>>>(xin, batch, sums, cnts);
  pool_div<<<(N_GRAPHS * DIM + 255) / 256, 256, 0, stream>>>(sums, cnts, (float*)d_out);
}